// net_need_name_80582176407954
// MI455X (gfx1250) — compile-verified
//
#include <hip/hip_runtime.h>

typedef __attribute__((ext_vector_type(2))) float v2f;
typedef __attribute__((ext_vector_type(8))) float v8f;

#define DIN  64
#define HF   128   // hidden features (GCN out)
#define DOUT 128
#define EPS  1e-5f

// ---------------------------------------------------------------------------
// GEMM1: h1[N x 128] = x[N x 64] @ W1[64 x 128]   (fp32 WMMA 16x16x4)
// block = 256 threads = 8 waves; wave w computes columns [16w, 16w+16)
// ---------------------------------------------------------------------------
__global__ __launch_bounds__(256) void k_gemm_xw1(const float* __restrict__ x,
                                                  const float* __restrict__ W1,
                                                  float* __restrict__ h1, int n) {
  const int lane = threadIdx.x & 31;
  const int wave = threadIdx.x >> 5;
  const int row0 = blockIdx.x * 16;
  const int m    = lane & 15;
  const int kh   = (lane >> 4) * 2;           // K-half offset: 0 or 2
  const int col  = wave * 16 + m;
  int rowA = row0 + m; if (rowA >= n) rowA = n - 1;
  const float* __restrict__ xr = x + (size_t)rowA * DIN;

  v8f acc = {};
  for (int k = 0; k < DIN; k += 4) {
    v2f a, b;
    a.x = xr[k + kh];
    a.y = xr[k + kh + 1];
    b.x = W1[(size_t)(k + kh)     * HF + col];
    b.y = W1[(size_t)(k + kh + 1) * HF + col];
    acc = __builtin_amdgcn_wmma_f32_16x16x4_f32(false, a, false, b,
                                                (short)0, acc, false, false);
  }
  const int rbase = row0 + (lane >> 4) * 8;
  for (int r = 0; r < 8; ++r) {
    int row = rbase + r;
    if (row < n) h1[(size_t)row * HF + col] = acc[r];
  }
}

// ---------------------------------------------------------------------------
// GEMM2 (fused SAGE): out[N x 128] = (agg * invcnt_row) @ Wl + h2 @ Wr
// (bl omitted: a constant row bias before training-mode BN cancels exactly)
// ---------------------------------------------------------------------------
__global__ __launch_bounds__(256) void k_gemm_sage(const float* __restrict__ agg,
                                                   const float* __restrict__ h2,
                                                   const float* __restrict__ Wl,
                                                   const float* __restrict__ Wr,
                                                   const float* __restrict__ invcnt,
                                                   float* __restrict__ out, int n) {
  const int lane = threadIdx.x & 31;
  const int wave = threadIdx.x >> 5;
  const int row0 = blockIdx.x * 16;
  const int m    = lane & 15;
  const int kh   = (lane >> 4) * 2;
  const int col  = wave * 16 + m;
  int rowA = row0 + m; if (rowA >= n) rowA = n - 1;
  const float rinv = invcnt[rowA];
  const float* __restrict__ ar = agg + (size_t)rowA * HF;
  const float* __restrict__ hr = h2  + (size_t)rowA * HF;

  v8f acc = {};
  for (int k = 0; k < HF; k += 4) {        // neighbor-mean branch
    v2f a, b;
    a.x = ar[k + kh]     * rinv;
    a.y = ar[k + kh + 1] * rinv;
    b.x = Wl[(size_t)(k + kh)     * DOUT + col];
    b.y = Wl[(size_t)(k + kh + 1) * DOUT + col];
    acc = __builtin_amdgcn_wmma_f32_16x16x4_f32(false, a, false, b,
                                                (short)0, acc, false, false);
  }
  for (int k = 0; k < HF; k += 4) {        // root branch
    v2f a, b;
    a.x = hr[k + kh];
    a.y = hr[k + kh + 1];
    b.x = Wr[(size_t)(k + kh)     * DOUT + col];
    b.y = Wr[(size_t)(k + kh + 1) * DOUT + col];
    acc = __builtin_amdgcn_wmma_f32_16x16x4_f32(false, a, false, b,
                                                (short)0, acc, false, false);
  }
  const int rbase = row0 + (lane >> 4) * 8;
  for (int r = 0; r < 8; ++r) {
    int row = rbase + r;
    if (row < n) out[(size_t)row * DOUT + col] = acc[r];
  }
}

// ---------------------------------------------------------------------------
// degree / normalization helpers
// ---------------------------------------------------------------------------
__global__ void k_indeg(const int* __restrict__ dst, float* __restrict__ indeg, int e) {
  int t = blockIdx.x * blockDim.x + threadIdx.x;
  if (t < e) atomicAdd(&indeg[dst[t]], 1.0f);
}

__global__ void k_dinv(const float* __restrict__ indeg, float* __restrict__ dinv,
                       float* __restrict__ invcnt, int n) {
  int t = blockIdx.x * blockDim.x + threadIdx.x;
  if (t < n) {
    float d = indeg[t];
    dinv[t]   = rsqrtf(d + 1.0f);                 // GCN: +1 self-loop
    invcnt[t] = 1.0f / (d > 1.0f ? d : 1.0f);     // SAGE mean divisor
  }
}

// gcn[i] = h1[i] * dinv[i]^2   (self-loop contribution, also initializes gcn)
__global__ void k_gcn_self(const float* __restrict__ h1, const float* __restrict__ dinv,
                           float* __restrict__ gcn, int n) {
  int t = blockIdx.x * blockDim.x + threadIdx.x;
  if (t < n * HF) {
    int i = t >> 7;                                // /128
    float w = dinv[i];
    gcn[t] = h1[t] * w * w;
  }
}

// per-(edge, 4-feature) scatter: gcn[dst] += h1[src] * dinv[src]*dinv[dst]
__global__ void k_gcn_edges(const float* __restrict__ h1, const int* __restrict__ src,
                            const int* __restrict__ dst, const float* __restrict__ dinv,
                            float* __restrict__ gcn, int e) {
  int t = blockIdx.x * blockDim.x + threadIdx.x;
  if (t >= e * 32) return;
  int ed = t >> 5;
  int f  = (t & 31) * 4;
  int s = src[ed], d = dst[ed];
  float w = dinv[s] * dinv[d];
  const float4 v = *(const float4*)(h1 + (size_t)s * HF + f);
  float* o = gcn + (size_t)d * HF + f;
  atomicAdd(o + 0, v.x * w);
  atomicAdd(o + 1, v.y * w);
  atomicAdd(o + 2, v.z * w);
  atomicAdd(o + 3, v.w * w);
}

// agg[dst] += h2[src]  (unweighted; mean divisor applied in GEMM2)
__global__ void k_sage_edges(const float* __restrict__ h2, const int* __restrict__ src,
                             const int* __restrict__ dst, float* __restrict__ agg, int e) {
  int t = blockIdx.x * blockDim.x + threadIdx.x;
  if (t >= e * 32) return;
  int ed = t >> 5;
  int f  = (t & 31) * 4;
  int s = src[ed], d = dst[ed];
  const float4 v = *(const float4*)(h2 + (size_t)s * HF + f);
  float* o = agg + (size_t)d * HF + f;
  atomicAdd(o + 0, v.x);
  atomicAdd(o + 1, v.y);
  atomicAdd(o + 2, v.z);
  atomicAdd(o + 3, v.w);
}

// ---------------------------------------------------------------------------
// BatchNorm (training mode, biased variance) over the node dimension
// ---------------------------------------------------------------------------
__global__ __launch_bounds__(128) void k_bn_stats(const float* __restrict__ in,
                                                  float* __restrict__ sum,
                                                  float* __restrict__ sq, int n) {
  const int c  = threadIdx.x;          // one column per thread, 128 cols
  const int r0 = blockIdx.x * 256;
  int rend = r0 + 256; if (rend > n) rend = n;
  float s = 0.0f, s2 = 0.0f;
  for (int r = r0; r < rend; ++r) {
    float v = in[(size_t)r * 128 + c]; // coalesced: 128 lanes -> 128 consecutive
    s += v; s2 += v * v;
  }
  atomicAdd(&sum[c], s);
  atomicAdd(&sq[c],  s2);
}

__global__ __launch_bounds__(128) void k_bn_finalize(const float* __restrict__ sum,
                                                     const float* __restrict__ sq,
                                                     const float* __restrict__ g,
                                                     const float* __restrict__ be,
                                                     float* __restrict__ scale,
                                                     float* __restrict__ shift, int n) {
  int c = threadIdx.x;
  float inv_n = 1.0f / (float)n;
  float mean = sum[c] * inv_n;
  float var  = sq[c] * inv_n - mean * mean;
  if (var < 0.0f) var = 0.0f;
  float s = g[c] * rsqrtf(var + EPS);
  scale[c] = s;
  shift[c] = be[c] - mean * s;
}

__global__ void k_bn_apply_relu(float* __restrict__ io, const float* __restrict__ scale,
                                const float* __restrict__ shift, int n) {
  int t = blockIdx.x * blockDim.x + threadIdx.x;
  if (t < n * 128) {
    int c = t & 127;
    float v = io[t] * scale[c] + shift[c];
    io[t] = v > 0.0f ? v : 0.0f;
  }
}

// ---------------------------------------------------------------------------
// launcher
// ---------------------------------------------------------------------------
extern "C" void kernel_launch(void* const* d_in, const int* in_sizes, int n_in,
                              void* d_out, int out_size, void* d_ws, size_t ws_size,
                              hipStream_t stream) {
  const float* x   = (const float*)d_in[0];
  const int*   src = (const int*)d_in[1];
  const int*   dst = (const int*)d_in[2];
  const float* W1  = (const float*)d_in[3];
  // d_in[4] = b1 (cancels in BN), d_in[5]=g1, d_in[6]=be1
  const float* g1  = (const float*)d_in[5];
  const float* be1 = (const float*)d_in[6];
  const float* Wl  = (const float*)d_in[7];
  // d_in[8] = bl (cancels in BN)
  const float* Wr  = (const float*)d_in[9];
  const float* g2  = (const float*)d_in[10];
  const float* be2 = (const float*)d_in[11];
  float* out = (float*)d_out;

  const int N = in_sizes[0] / DIN;
  const int E = in_sizes[1];
  const size_t NF = (size_t)N * HF;

  // workspace carve (base is 256B-aligned; all chunks keep 16B alignment)
  char* p = (char*)d_ws;
  float* h1     = (float*)p; p += NF * sizeof(float);      // x @ W1
  float* gcn    = (float*)p; p += NF * sizeof(float);      // GCN out -> h2 in place
  float* agg    = (float*)p; p += NF * sizeof(float);      // SAGE neighbor sum
  float* indeg  = (float*)p; p += (size_t)N * sizeof(float);
  float* dinv   = (float*)p; p += (size_t)N * sizeof(float);
  float* invcnt = (float*)p; p += (size_t)N * sizeof(float);
  float* stats  = (float*)p;                               // 1024 floats
  float *sum1 = stats,        *sq1 = stats + 128, *sc1 = stats + 256, *sh1 = stats + 384;
  float *sum2 = stats + 512,  *sq2 = stats + 640, *sc2 = stats + 768, *sh2 = stats + 896;

  // zero accumulation buffers (memset nodes are graph-capturable)
  hipMemsetAsync(indeg, 0, (size_t)N * sizeof(float), stream);
  hipMemsetAsync(agg,   0, NF * sizeof(float), stream);
  hipMemsetAsync(stats, 0, 1024 * sizeof(float), stream);

  const int tilesN   = (N + 15) / 16;
  const int elemBlk  = (int)((NF + 255) / 256);
  const int edgeBlk  = (E + 255) / 256;
  const int edgeQBlk = (int)(((size_t)E * 32 + 255) / 256);
  const int statBlk  = (N + 255) / 256;

  // --- layer 1: GCNConv ---
  k_gemm_xw1 <<<tilesN, 256, 0, stream>>>(x, W1, h1, N);
  k_indeg    <<<edgeBlk, 256, 0, stream>>>(dst, indeg, E);
  k_dinv     <<<(N + 255) / 256, 256, 0, stream>>>(indeg, dinv, invcnt, N);
  k_gcn_self <<<elemBlk, 256, 0, stream>>>(h1, dinv, gcn, N);
  k_gcn_edges<<<edgeQBlk, 256, 0, stream>>>(h1, src, dst, dinv, gcn, E);

  // --- BN1 + ReLU (in place on gcn -> h2) ---
  k_bn_stats   <<<statBlk, 128, 0, stream>>>(gcn, sum1, sq1, N);
  k_bn_finalize<<<1, 128, 0, stream>>>(sum1, sq1, g1, be1, sc1, sh1, N);
  k_bn_apply_relu<<<elemBlk, 256, 0, stream>>>(gcn, sc1, sh1, N);

  // --- layer 2: SAGEConv ---
  k_sage_edges<<<edgeQBlk, 256, 0, stream>>>(gcn, src, dst, agg, E);
  k_gemm_sage <<<tilesN, 256, 0, stream>>>(agg, gcn, Wl, Wr, invcnt, out, N);

  // --- BN2 + ReLU (in place on out) ---
  k_bn_stats   <<<statBlk, 128, 0, stream>>>(out, sum2, sq2, N);
  k_bn_finalize<<<1, 128, 0, stream>>>(sum2, sq2, g2, be2, sc2, sh2, N);
  k_bn_apply_relu<<<elemBlk, 256, 0, stream>>>(out, sc2, sh2, N);
}